// Loss_59811714564490
// MI455X (gfx1250) — compile-verified
//
#include <hip/hip_runtime.h>

// ---------------- problem constants (from reference) ----------------
#define NCLS 20
#define BSZ 256
#define HH  26
#define WW  26
#define AA  5
#define HW  (HH * WW)          // 676
#define CPA (5 + NCLS)         // 25 channels per anchor
#define PRED_B_STRIDE (AA * CPA * HW)  // 125*676 per batch item

__constant__ float c_anch_w[5] = {1.3221f, 3.19275f, 5.05587f, 9.47112f, 11.2364f};
__constant__ float c_anch_h[5] = {1.73145f, 4.00944f, 8.09892f, 4.84053f, 10.0071f};

#define LAMBDA_COORD 5.0f
#define LAMBDA_OBJ   1.0f
#define LAMBDA_NOOBJ 0.5f
#define LAMBDA_CLS   1.0f

typedef __attribute__((ext_vector_type(2))) float v2f;
typedef __attribute__((ext_vector_type(8))) float v8f;

// ---------------- full 32-lane f32 sum via V_WMMA_F32_16X16X4_F32 ----------------
// Stage 1: A carries per-lane values (A[m][0]=val(lane m), A[m][2]=val(lane m+16),
//          A[m][1]=A[m][3]=0), B = ones(4x16)  =>  R[m][n] = val(m)+val(m+16).
// Stage 2: A = ones(16x4); B built from pairs of R's D-VGPRs:
//          pair (d_{2j}, d_{2j+1}) supplies columns {r_{2j}, r_{2j+8}, r_{2j+1}, r_{2j+9}}.
//          Four accumulating WMMAs sum all 16 row-sums => total broadcast to all lanes.
// Requires EXEC all-1s (guaranteed: exact grid, no divergence at call site).
__device__ __forceinline__ float wave_sum32(float v) {
    v2f a;   a[0] = v;    a[1] = 0.0f;
    v2f one; one[0] = 1.0f; one[1] = 1.0f;
    v8f z = {};
    v8f r = __builtin_amdgcn_wmma_f32_16x16x4_f32(
        /*neg_a=*/false, a, /*neg_b=*/false, one,
        /*c_mod=*/(short)0, z, /*reuse_a=*/false, /*reuse_b=*/false);
    v8f acc = {};
#pragma unroll
    for (int j = 0; j < 4; ++j) {
        v2f bb; bb[0] = r[2 * j]; bb[1] = r[2 * j + 1];
        acc = __builtin_amdgcn_wmma_f32_16x16x4_f32(
            false, one, false, bb, (short)0, acc, false, false);
    }
    return acc[0];
}

__device__ __forceinline__ float sigmoidf_(float x) {
    return 1.0f / (1.0f + __expf(-x));
}

__global__ void zero_out4(float* __restrict__ out) {
    if (threadIdx.x < 4) out[threadIdx.x] = 0.0f;
}

// One thread per (b, h, w) cell. 676 blocks * 256 threads == 256*26*26 exactly.
__global__ __launch_bounds__(256) void yolo_loss_kernel(
    const float* __restrict__ pred,   // (B, 125, H, W)
    const float* __restrict__ tgt,    // (B, H, W, 25)
    float* __restrict__ out)          // 4 scalars
{
    const int i  = blockIdx.x * blockDim.x + threadIdx.x;  // cell id
    const int b  = i / HW;
    const int hw = i - b * HW;

    // ---- target: always read the 5 box/conf fields; class one-hot only if obj ----
    const float* t = tgt + (size_t)i * CPA;
    const float gconf = t[20];
    const float gx = t[21], gy = t[22], gw = t[23], gh = t[24];
    const bool  obj  = (gconf != 0.0f);
    const float fobj = obj ? 1.0f : 0.0f;

    const float gx1 = gx - 0.5f * gw, gy1 = gy - 0.5f * gh;
    const float gx2 = gx + 0.5f * gw, gy2 = gy + 0.5f * gh;
    const float garea = gw * gh;

    // ---- prediction: only channels 20..24 of each anchor (skip 100 cls channels) ----
    const float* p = pred + (size_t)b * PRED_B_STRIDE + hw;

    float best_iou = -1.0f;
    int   best_a   = 0;
    float bx = 0.f, by = 0.f, bw = 0.f, bh = 0.f, bc = 0.f;

#pragma unroll
    for (int a = 0; a < AA; ++a) {
        const float* pa = p + (size_t)(a * CPA) * HW;
        const float sc = sigmoidf_(pa[20 * HW]);
        const float sx = sigmoidf_(pa[21 * HW]);
        const float sy = sigmoidf_(pa[22 * HW]);
        const float ew = __expf(pa[23 * HW]) * c_anch_w[a];
        const float eh = __expf(pa[24 * HW]) * c_anch_h[a];

        const float ax1 = sx - 0.5f * ew, ay1 = sy - 0.5f * eh;
        const float ax2 = sx + 0.5f * ew, ay2 = sy + 0.5f * eh;
        const float iw  = fmaxf(fminf(ax2, gx2) - fmaxf(ax1, gx1), 0.0f);
        const float ih  = fmaxf(fminf(ay2, gy2) - fmaxf(ay1, gy1), 0.0f);
        const float inter = iw * ih;
        const float uni   = ew * eh + garea - inter;
        const float iou   = inter / (uni + 1e-10f);

        if (iou > best_iou) {  // strict > == first-max, matches jnp.argmax
            best_iou = iou; best_a = a;
            bx = sx; by = sy; bw = ew; bh = eh; bc = sc;
        }
    }

    // ---- losses ----
    const float dx = bx - gx, dy = by - gy, dw = bw - gw, dh = bh - gh;
    float box_l  = fobj * (dx * dx + dy * dy + dw * dw + dh * dh);
    const float dc = bc - gconf;
    float conf_l = fobj * dc * dc;
    float noob_l = (1.0f - fobj) * bc * bc;
    float cls_l  = 0.0f;

    if (obj) {
        // label = argmax of one-hot (first max)
        int label = 0; float lm = t[0];
#pragma unroll
        for (int c = 1; c < NCLS; ++c) {
            const float v = t[c];
            if (v > lm) { lm = v; label = c; }
        }
        // class logits of the best anchor only (20 coalesced-within-active-lanes loads)
        const float* pa = p + (size_t)(best_a * CPA) * HW;
        float lg[NCLS];
        float mx = -3.402823466e38f;
        float ll = 0.0f;
#pragma unroll
        for (int c = 0; c < NCLS; ++c) {
            const float v = pa[c * HW];
            lg[c] = v;
            mx = fmaxf(mx, v);
            if (c == label) ll = v;
        }
        float s = 0.0f;
#pragma unroll
        for (int c = 0; c < NCLS; ++c) s += __expf(lg[c] - mx);
        const float lse = mx + __logf(s);
        cls_l = lse - ll;  // -log_softmax[label]
    }

    // fold lambda / bsize scaling into the partials
    const float inv_b = 1.0f / (float)BSZ;
    box_l  *= LAMBDA_COORD * inv_b;
    conf_l *= LAMBDA_OBJ   * inv_b;
    noob_l *= LAMBDA_NOOBJ * inv_b;
    cls_l  *= LAMBDA_CLS   * inv_b;

    // ---- wave-level reduction on the matrix unit (EXEC all-1s here) ----
    box_l  = wave_sum32(box_l);
    conf_l = wave_sum32(conf_l);
    noob_l = wave_sum32(noob_l);
    cls_l  = wave_sum32(cls_l);

    // ---- block combine (8 wave32s) + global atomic ----
    __shared__ float ls[8][4];
    const int wid  = threadIdx.x >> 5;
    const int lane = threadIdx.x & 31;
    if (lane == 0) {
        ls[wid][0] = box_l; ls[wid][1] = conf_l;
        ls[wid][2] = noob_l; ls[wid][3] = cls_l;
    }
    __syncthreads();
    if (threadIdx.x < 4) {
        float s = 0.0f;
#pragma unroll
        for (int wv = 0; wv < 8; ++wv) s += ls[wv][threadIdx.x];
        atomicAdd(&out[threadIdx.x], s);
    }
}

extern "C" void kernel_launch(void* const* d_in, const int* in_sizes, int n_in,
                              void* d_out, int out_size, void* d_ws, size_t ws_size,
                              hipStream_t stream) {
    (void)in_sizes; (void)n_in; (void)out_size; (void)d_ws; (void)ws_size;
    const float* pred = (const float*)d_in[0];
    const float* tgt  = (const float*)d_in[1];
    float* out = (float*)d_out;

    zero_out4<<<1, 32, 0, stream>>>(out);
    // 676 blocks * 256 threads == 256*26*26 cells exactly (EXEC all-1s for WMMA)
    yolo_loss_kernel<<<HW, 256, 0, stream>>>(pred, tgt, out);
}